// HierSoftmax_3917010174471
// MI455X (gfx1250) — compile-verified
//
#include <hip/hip_runtime.h>

// Problem constants (match reference)
#define VOCAB      50257
#define DMODEL     512
#define MAX_DEPTH  20
#define NTOK       8192   // B * S = 8 * 1024

typedef __attribute__((ext_vector_type(2))) float v2f;
typedef __attribute__((ext_vector_type(8))) float v8f;

__device__ __forceinline__ float log_sigmoid_stable(float x) {
    // log(sigmoid(x)) = min(x, 0) - log1p(exp(-|x|))   (numerically stable)
    return fminf(x, 0.0f) - log1pf(__expf(-fabsf(x)));
}

// One wave32 per token. Per token:
//   D-tile = A(16x4 f32, gathered embedding rows) x B(4x16 f32, hidden chunk
//   replicated across columns) accumulated over K=512 via V_WMMA_F32_16X16X4_F32.
// Two M-tiles cover depths 0..15 and 16..19 (rows >= MAX_DEPTH use row 0 of
// node_emb and are masked out by the d < len test in the epilogue).
__global__ __launch_bounds__(256) void hier_softmax_nll_kernel(
    const float* __restrict__ hidden,      // [NTOK, DMODEL]
    const float* __restrict__ node_emb,    // [VOCAB-1, DMODEL]
    const float* __restrict__ path_signs,  // [VOCAB, MAX_DEPTH]
    const int*   __restrict__ targets,     // [NTOK]
    const int*   __restrict__ path_nodes,  // [VOCAB, MAX_DEPTH]
    const int*   __restrict__ path_lens,   // [VOCAB]
    float*       __restrict__ out)         // [1] scalar mean NLL
{
    __shared__ __align__(16) float lds_h[8 * DMODEL];   // one h-vector per wave

    const int lane  = threadIdx.x & 31;
    const int wslot = threadIdx.x >> 5;
    const int wavesPerBlock = blockDim.x >> 5;
    const int gwave  = blockIdx.x * wavesPerBlock + wslot;
    const int nwaves = gridDim.x * wavesPerBlock;

    const int half  = lane >> 4;        // 0: lanes 0-15, 1: lanes 16-31
    const int mrow  = lane & 15;        // M row owned by this lane (A operand)
    const int klane = half * 2;         // K sub-offset within a 4-wide K step

    float* hl = &lds_h[wslot * DMODEL];
    float local_sum = 0.0f;

    for (int t = gwave; t < NTOK; t += nwaves) {
        // ---- stage hidden[t] (512 f32) into this wave's LDS slot ----
        const float4* hsrc = (const float4*)(hidden + (size_t)t * DMODEL);
        float4*       hdst = (float4*)hl;
        #pragma unroll
        for (int c = 0; c < 4; ++c)
            hdst[lane + c * 32] = hsrc[lane + c * 32];   // 128 float4 total
        // (intra-wave LDS ops are in-order; no barrier needed, LDS slot is wave-private)

        const int tt = targets[t];
        int len = path_lens[tt];
        len = (len < 1) ? 1 : len;       // depth >= 1, and len <= 19 < MAX_DEPTH

        // A-operand rows: gathered embedding rows for this token's path
        const int node0 = path_nodes[tt * MAX_DEPTH + mrow];            // depths 0..15
        const int row1  = 16 + mrow;
        const int node1 = (row1 < MAX_DEPTH)
                              ? path_nodes[tt * MAX_DEPTH + row1] : 0;  // depths 16..19
        const float* w0 = node_emb + (size_t)node0 * DMODEL;
        const float* w1 = node_emb + (size_t)node1 * DMODEL;

        v8f acc0 = {};   // logits for depths  0..15 (replicated across N columns)
        v8f acc1 = {};   // logits for depths 16..31 (only 16..19 meaningful)

        #pragma unroll 4
        for (int k = 0; k < DMODEL; k += 4) {
            // B 4x16 layout: lane half selects K base (0 or 2), VGPR j adds j.
            v2f b  = *(const v2f*)(hl + k + klane);     // ds_load_b64 (same for A & B k-offset)
            v2f a0 = *(const v2f*)(w0 + k + klane);     // global_load_b64, 256 B/step of real data
            v2f a1 = *(const v2f*)(w1 + k + klane);
            acc0 = __builtin_amdgcn_wmma_f32_16x16x4_f32(
                       false, a0, false, b, (short)0, acc0, false, false);
            acc1 = __builtin_amdgcn_wmma_f32_16x16x4_f32(
                       false, a1, false, b, (short)0, acc1, false, false);
        }

        // ---- epilogue: C layout -> VGPR v holds row M = v + 8*half ----
        float part = 0.0f;
        #pragma unroll
        for (int v = 0; v < 8; ++v) {
            const int d0 = half * 8 + v;               // depth from tile 0
            if (d0 < len) {
                float sg = (path_signs[tt * MAX_DEPTH + d0] >= 0.0f) ? 1.0f : -1.0f;
                part += log_sigmoid_stable(sg * acc0[v]);
            }
            const int d1 = 16 + half * 8 + v;          // depth from tile 1
            if (d1 < MAX_DEPTH && d1 < len) {
                float sg = (path_signs[tt * MAX_DEPTH + d1] >= 0.0f) ? 1.0f : -1.0f;
                part += log_sigmoid_stable(sg * acc1[v]);
            }
        }
        // lanes 0-15 all hold identical 'part' (depths 0-7 & 16-19);
        // lanes 16-31 all hold identical 'part' (depths 8-15).
        const float p_lo = __shfl(part, 0, 32);
        const float p_hi = __shfl(part, 16, 32);
        local_sum += -(p_lo + p_hi);                   // token NLL
    }

    if (lane == 0 && local_sum != 0.0f)
        atomicAdd(out, local_sum * (1.0f / (float)NTOK));
    else if (lane == 0)
        atomicAdd(out, local_sum * (1.0f / (float)NTOK));
}

extern "C" void kernel_launch(void* const* d_in, const int* in_sizes, int n_in,
                              void* d_out, int out_size, void* d_ws, size_t ws_size,
                              hipStream_t stream) {
    (void)in_sizes; (void)n_in; (void)d_ws; (void)ws_size; (void)out_size;

    const float* hidden     = (const float*)d_in[0];
    const float* node_emb   = (const float*)d_in[1];
    const float* path_signs = (const float*)d_in[2];
    const int*   targets    = (const int*)d_in[3];
    const int*   path_nodes = (const int*)d_in[4];
    const int*   path_lens  = (const int*)d_in[5];
    float*       out        = (float*)d_out;

    // d_out is poisoned; zero the scalar accumulator (memset node is graph-capturable)
    hipMemsetAsync(out, 0, sizeof(float), stream);

    // 256 blocks x 8 waves = 2048 waves; 4 tokens per wave, uniform trip counts.
    dim3 grid(256), block(256);
    hier_softmax_nll_kernel<<<grid, block, 0, stream>>>(
        hidden, node_emb, path_signs, targets, path_nodes, path_lens, out);
}